// STU_2929167696035
// MI455X (gfx1250) — compile-verified
//
#include <hip/hip_runtime.h>
#include <math.h>

// Problem constants (match reference)
#define NB 2
#define NL 2048
#define ND 512
#define NK 16
#define NT 13
#define NFFT 4096
#define FH 2049        // rfft spectrum length
#define APITCH 516     // LDS row pitch (floats) = 512 + 2 TDM pads of 2 DWORDs

typedef __attribute__((ext_vector_type(2))) float v2f;
typedef __attribute__((ext_vector_type(8))) float v8f;
typedef __attribute__((ext_vector_type(4))) unsigned int v4u;
typedef __attribute__((ext_vector_type(8))) int v8i;
typedef __attribute__((ext_vector_type(4))) int v4i;

static __device__ __forceinline__ int brev12(int x) {
  return (int)(__brev((unsigned)x) >> 20);
}

// TDM-padded LDS mapping: linear (row*512+col) with 2 DWORDs of pad inserted
// after every 256 DWORDs -> addr = row*516 + (col>>8)*2 + col
static __device__ __forceinline__ int lds_off(int row, int col) {
  return row * APITCH + ((col >> 8) << 1) + col;
}

// ---------------------------------------------------------------------------
// K0: QeffT[s][d][r] = sum_i p[i] * Q[clip(i+1-13+s,0,12)][r][d]
// Transposed output so the WMMA B fragment is a single contiguous b64 load.
// LDS 32x33 tile transpose keeps both global read and write coalesced.
// ---------------------------------------------------------------------------
__global__ void k_qeff_t(const float* __restrict__ Q, const float* __restrict__ p,
                         float* __restrict__ QeffT) {
  __shared__ float tile[32][33];
  const int SZ = ND * ND;
  int s = blockIdx.z;
  int r0 = blockIdx.y * 32;
  int d0 = blockIdx.x * 32;
  int lx = threadIdx.x & 31;
  int ly = threadIdx.x >> 5;   // 0..7
  float pc[NT];
#pragma unroll
  for (int i = 0; i < NT; ++i) pc[i] = p[i];
#pragma unroll
  for (int pass = 0; pass < 4; ++pass) {
    int j = ly + pass * 8;     // r offset in tile
    float acc = 0.f;
#pragma unroll
    for (int i = 0; i < NT; ++i) {
      int off = i + 1 - NT + s;
      off = off < 0 ? 0 : (off > NT - 1 ? NT - 1 : off);
      acc += pc[i] * Q[(size_t)off * SZ + (size_t)(r0 + j) * ND + d0 + lx];
    }
    tile[j][lx] = acc;
  }
  __syncthreads();
#pragma unroll
  for (int pass = 0; pass < 4; ++pass) {
    int j = ly + pass * 8;     // d offset in tile
    QeffT[(size_t)s * SZ + (size_t)(d0 + j) * ND + r0 + lx] = tile[lx][j];
  }
}

// ---------------------------------------------------------------------------
// K0b: Msum[k,i] = sum_d M[d][k][i]   (einsum 'blki,dki->bli' sums d out of M)
// ---------------------------------------------------------------------------
__global__ void k_msum(const float* __restrict__ M, float* __restrict__ Msum) {
  int e = blockIdx.x * blockDim.x + threadIdx.x; // e = k*ND + i
  if (e >= NK * ND) return;
  float acc = 0.f;
  for (int d = 0; d < ND; ++d) acc += M[(size_t)d * (NK * ND) + e];
  Msum[e] = acc;
}

// ---------------------------------------------------------------------------
// In-LDS radix-2 DIT FFT, length 4096, input already bit-reversed.
// sign = -1 forward, +1 inverse (unscaled).
// ---------------------------------------------------------------------------
static __device__ void fft4096(float2* s, int tid, int nthr, float sign) {
  for (int st = 0; st < 12; ++st) {
    __syncthreads();
    int half = 1 << st;
    for (int t = tid; t < NFFT / 2; t += nthr) {
      int pos = t & (half - 1);
      int i0 = ((t >> st) << (st + 1)) + pos;
      int i1 = i0 + half;
      float ang = sign * 3.14159265358979323846f * (float)pos / (float)half;
      float sw, cw;
      __sincosf(ang, &sw, &cw);
      float2 a = s[i0], b = s[i1];
      float2 tw = make_float2(cw * b.x - sw * b.y, cw * b.y + sw * b.x);
      s[i0] = make_float2(a.x + tw.x, a.y + tw.y);
      s[i1] = make_float2(a.x - tw.x, a.y - tw.y);
    }
  }
  __syncthreads();
}

// ---------------------------------------------------------------------------
// K1: U[b,:,i] = FFT_4096(zero-pad u[b,:,i]); one workgroup per (b,i)
// ---------------------------------------------------------------------------
__global__ void k_fft_u(const float* __restrict__ u, float2* __restrict__ Uf) {
  __shared__ float2 s[NFFT];
  int bid = blockIdx.x;
  int b = bid >> 9, i = bid & (ND - 1);
  for (int p = threadIdx.x; p < NFFT; p += blockDim.x) {
    int n = brev12(p);
    float re = (n < NL) ? u[((size_t)b * NL + n) * ND + i] : 0.f;
    s[p] = make_float2(re, 0.f);
  }
  fft4096(s, threadIdx.x, blockDim.x, -1.f);
  for (int p = threadIdx.x; p < NFFT; p += blockDim.x)
    Uf[(size_t)bid * NFFT + p] = s[p];
}

// ---------------------------------------------------------------------------
// K1b: Vf[k][f] = FFT_4096(zero-pad filters[:,k]); one workgroup per k
// ---------------------------------------------------------------------------
__global__ void k_fft_filters(const float* __restrict__ filters,
                              float2* __restrict__ Vf) {
  __shared__ float2 s[NFFT];
  int k = blockIdx.x;
  for (int p = threadIdx.x; p < NFFT; p += blockDim.x) {
    int n = brev12(p);
    float re = (n < NL) ? filters[(size_t)n * NK + k] : 0.f;
    s[p] = make_float2(re, 0.f);
  }
  fft4096(s, threadIdx.x, blockDim.x, -1.f);
  for (int p = threadIdx.x; p < NFFT; p += blockDim.x)
    Vf[(size_t)k * NFFT + p] = s[p];
}

// ---------------------------------------------------------------------------
// K2: W[i][f] = sum_k Msum[k,i] * Vf[k][f]   (f in [0, FH))
// ---------------------------------------------------------------------------
__global__ void k_w(const float2* __restrict__ Vf, const float* __restrict__ Msum,
                    float2* __restrict__ W) {
  int f = blockIdx.x * blockDim.x + threadIdx.x;
  int i = blockIdx.y;
  if (f >= FH) return;
  float2 acc = make_float2(0.f, 0.f);
#pragma unroll
  for (int k = 0; k < NK; ++k) {
    float m = Msum[k * ND + i];
    float2 v = Vf[(size_t)k * NFFT + f];
    acc.x += m * v.x;
    acc.y += m * v.y;
  }
  W[(size_t)i * FH + f] = acc;
}

// ---------------------------------------------------------------------------
// K4: out[b,t,d] = bi + uni, bi via V_WMMA_F32_16X16X4_F32.
// A panel (28 rows of u, t0-12..t0+15) staged in LDS by the Tensor Data Mover
// (tensor_load_to_lds issued from wave 0, D# with pad_enable reproducing the
// 516-float anti-bank-conflict pitch), completion via s_wait_tensorcnt + bar.
// B fragments are single b64 loads from the transposed QeffT[s][d][r].
// A frag layout (ISA 7.12.2): lanes 0-15 -> M=lane, K={0,1}; lanes 16-31 ->
// M=lane-16, K={2,3}. B mirrored with N in place of M. C: VGPR j -> rows j /
// j+8 for lane halves, N = lane&15.
// ---------------------------------------------------------------------------
__global__ void __launch_bounds__(256)
k_biuni(const float* __restrict__ u, const float* __restrict__ QeffT,
        const float* __restrict__ p, float* __restrict__ out) {
  __shared__ float Apan[28 * APITCH];
  int rt = blockIdx.y;
  int b = rt >> 7;
  int t0 = (rt & 127) << 4;

  int gstart = t0 - 12;
  int skip = (gstart < 0) ? -gstart : 0;  // rows before t=0 (first tile only)
  int nrows = 28 - skip;

  // zero-fill the pre-history rows (disjoint from the TDM destination)
  for (int idx = threadIdx.x; idx < skip * ND; idx += blockDim.x) {
    int j = idx >> 9, r = idx & (ND - 1);
    Apan[lds_off(j, r)] = 0.f;
  }

  if ((threadIdx.x >> 5) == 0) {  // wave 0 drives the TDM
    unsigned lds_byte = (unsigned)(uintptr_t)(void*)&Apan[0] +
                        (unsigned)(lds_off(skip, 0) * 4);
    unsigned long long ga = (unsigned long long)(uintptr_t)(
        u + ((size_t)b * NL + (gstart + skip)) * ND);
    v4u g0v;
    g0v[0] = 1u;                                    // count=1, user descriptor
    g0v[1] = lds_byte;                              // lds_addr
    g0v[2] = (unsigned)(ga & 0xffffffffu);          // global_addr[31:0]
    g0v[3] = (unsigned)((ga >> 32) & 0x01ffffffu)   // global_addr[56:32]
             | (2u << 30);                          // type = 2 ("image")
    v8i g1v;
    g1v[0] = (int)((2u << 16)     // data_size = 4 bytes
                   | (1u << 20)   // pad_enable
                   | (7u << 22)   // pad_interval: every 256 DWORDs
                   | (1u << 25)); // pad_amount: 2 DWORDs
    g1v[1] = (int)(512u << 16);   // tensor_dim0 = 512  (bits 79:48)
    g1v[2] = (int)((unsigned)nrows << 16); // tensor_dim1 = nrows (bits 111:80)
    g1v[3] = (int)(512u << 16);   // tile_dim0 = 512    (bits 127:112)
    g1v[4] = nrows;               // tile_dim1          (bits 143:128)
    g1v[5] = 512;                 // tensor_dim0_stride (bits 207:160)
    g1v[6] = 0;
    g1v[7] = 0;
    v4i zz4 = {0, 0, 0, 0};
    v8i zz8 = {0, 0, 0, 0, 0, 0, 0, 0};
    // amdgpu-toolchain (clang-23) 6-arg form:
    // (g0 v4u, g1 v8i, g2 v4i, g3 v4i, extra v8i, cpol i32)
    __builtin_amdgcn_tensor_load_to_lds(g0v, g1v, zz4, zz4, zz8, 0);
    __builtin_amdgcn_s_wait_tensorcnt(0);
  }
  __syncthreads();

  int wave = threadIdx.x >> 5;
  int lane = threadIdx.x & 31;
  int hi = lane >> 4;     // 0: K-pair {0,1}, 1: K-pair {2,3}
  int ln = lane & 15;     // M (for A) / N (for B, C)
  int d0 = blockIdx.x * 128 + wave * 16;

  v8f acc = {0.f, 0.f, 0.f, 0.f, 0.f, 0.f, 0.f, 0.f};

  for (int sft = 0; sft < NT; ++sft) {
    const float* qs =
        QeffT + (size_t)sft * ND * ND + (size_t)(d0 + ln) * ND + hi * 2;
    if (sft + 1 < NT)
      __builtin_prefetch(qs + (size_t)ND * ND, 0, 0);  // global_prefetch_b8
    int rowA = (12 - sft + ln) * APITCH + hi * 2;
    for (int r0 = 0; r0 < ND; r0 += 4) {
      v2f a;
      const float* ap = &Apan[rowA + r0 + ((r0 >> 8) << 1)];
      a.x = ap[0];
      a.y = ap[1];
      v2f bm = *(const v2f*)(qs + r0);  // contiguous pair -> global_load_b64
      acc = __builtin_amdgcn_wmma_f32_16x16x4_f32(false, a, false, bm,
                                                  (short)0, acc, false, false);
    }
  }

  // uni: out -= sum_s v[s]*u[b,t-s,d], v[s] = p_coeffs[12-s]
  float vc[NT];
#pragma unroll
  for (int sft = 0; sft < NT; ++sft) vc[sft] = p[12 - sft];

#pragma unroll
  for (int j = 0; j < 8; ++j) {
    int m = j + hi * 8;
    float uni = 0.f;
#pragma unroll
    for (int sft = 0; sft < NT; ++sft)
      uni += vc[sft] * Apan[lds_off(m + 12 - sft, d0 + ln)];
    float r = acc[j] - uni;
    out[((size_t)b * NL + t0 + m) * ND + d0 + ln] = r;
  }
}

// ---------------------------------------------------------------------------
// K3: spec: G[f] = Wfull[f] * U[b,i,f] (Hermitian extension), inverse FFT,
// out[b,t,i] += Re(g[t]) / NFFT. One workgroup per (b,i).
// ---------------------------------------------------------------------------
__global__ void k_spec(const float2* __restrict__ Uf, const float2* __restrict__ W,
                       float* __restrict__ out) {
  __shared__ float2 s[NFFT];
  int bid = blockIdx.x;
  int b = bid >> 9, i = bid & (ND - 1);
  const float2* ub = Uf + (size_t)bid * NFFT;
  const float2* wi = W + (size_t)i * FH;
  for (int p = threadIdx.x; p < NFFT; p += blockDim.x) {
    int n = brev12(p);
    float2 g;
    if (n <= NFFT / 2) {
      float2 w = wi[n];
      float2 uv = ub[n];
      g = make_float2(w.x * uv.x - w.y * uv.y, w.x * uv.y + w.y * uv.x);
    } else {
      int m2 = NFFT - n;
      float2 w = wi[m2];
      float2 uv = ub[m2];
      g = make_float2(w.x * uv.x - w.y * uv.y, -(w.x * uv.y + w.y * uv.x));
    }
    s[p] = g;
  }
  fft4096(s, threadIdx.x, blockDim.x, +1.f);
  const float inv = 1.f / (float)NFFT;
  for (int t = threadIdx.x; t < NL; t += blockDim.x)
    out[((size_t)b * NL + t) * ND + i] += s[t].x * inv;
}

// ---------------------------------------------------------------------------
// Launch: inputs in setup_inputs() order: u, filters, Q, M, p_coeffs (all f32).
// Workspace (floats): QeffT | Msum | Uf | Vf | W  (~56 MB total).
// ---------------------------------------------------------------------------
extern "C" void kernel_launch(void* const* d_in, const int* in_sizes, int n_in,
                              void* d_out, int out_size, void* d_ws, size_t ws_size,
                              hipStream_t stream) {
  (void)in_sizes; (void)n_in; (void)out_size; (void)ws_size;
  const float* u = (const float*)d_in[0];
  const float* filters = (const float*)d_in[1];
  const float* Q = (const float*)d_in[2];
  const float* M = (const float*)d_in[3];
  const float* p = (const float*)d_in[4];
  float* out = (float*)d_out;
  float* ws = (float*)d_ws;

  size_t o = 0;
  float* QeffT = ws + o; o += (size_t)NT * ND * ND;
  float* Msum = ws + o; o += (size_t)NK * ND;
  o = (o + 1) & ~(size_t)1;  // float2 alignment
  float2* Uf = (float2*)(ws + o); o += (size_t)NB * ND * NFFT * 2;
  float2* Vf = (float2*)(ws + o); o += (size_t)NK * NFFT * 2;
  float2* W  = (float2*)(ws + o); o += (size_t)ND * FH * 2;

  dim3 gq(ND / 32, ND / 32, NT);
  k_qeff_t<<<gq, 256, 0, stream>>>(Q, p, QeffT);
  k_msum<<<(NK * ND + 255) / 256, 256, 0, stream>>>(M, Msum);
  k_fft_u<<<NB * ND, 256, 0, stream>>>(u, Uf);
  k_fft_filters<<<NK, 256, 0, stream>>>(filters, Vf);
  dim3 gw((FH + 255) / 256, ND);
  k_w<<<gw, 256, 0, stream>>>(Vf, Msum, W);
  dim3 gb(4, (NB * NL) / 16);
  k_biuni<<<gb, 256, 0, stream>>>(u, QeffT, p, out);  // writes out (uni+bi)
  k_spec<<<NB * ND, 256, 0, stream>>>(Uf, W, out);    // += spec
}